// GNN_56856777064844
// MI455X (gfx1250) — compile-verified
//
#include <hip/hip_runtime.h>
#include <math.h>

#define N_NODES  50000
#define N_EDGES  800000
#define D        96
#define N_GRAPHS 50

typedef float v2f __attribute__((ext_vector_type(2)));
typedef float v8f __attribute__((ext_vector_type(8)));

// -------- float atomic max via int punning (portable, branch on sign) --------
__device__ __forceinline__ void atomicMaxF(float* addr, float v) {
    if (v >= 0.0f) atomicMax((int*)addr, __float_as_int(v));
    else           atomicMin((unsigned int*)addr, __float_as_uint(v));
}

// -------- init workspace: segmax=-inf, zeros elsewhere --------
__global__ void GNN_init(float* segmax, float* denom, float* aggr,
                         float* sums, float* cnt) {
    unsigned t = blockIdx.x * blockDim.x + threadIdx.x;
    if (t < (unsigned)(N_NODES * D)) aggr[t] = 0.0f;
    if (t < (unsigned)N_NODES) { segmax[t] = -INFINITY; denom[t] = 0.0f; }
    if (t < (unsigned)(N_GRAPHS * D)) sums[t] = 0.0f;
    if (t < (unsigned)N_GRAPHS) cnt[t] = 0.0f;
}

// -------- fused 4-weight GEMM: t{1..4} = x @ w{1..4} + b{1..4} --------
// One wave per 16x16 output tile. V_WMMA_F32_16X16X4_F32, K stepped by 4.
// A-frag (16x4, ISA layout): lanes 0-15 hold K=k0,k0+1; lanes 16-31 K=k0+2,k0+3.
// B-frag (4x16): mirrored row striping across lanes within each VGPR.
// C/D (16x16 f32): VGPR r, lane L -> row r + 8*(L/16), col L%16.
__global__ void GNN_gemm4_wmma(const float* __restrict__ x,
    const float* __restrict__ w1, const float* __restrict__ b1,
    const float* __restrict__ w2, const float* __restrict__ b2,
    const float* __restrict__ w3, const float* __restrict__ b3,
    const float* __restrict__ w4, const float* __restrict__ b4,
    float* __restrict__ t1, float* __restrict__ t2,
    float* __restrict__ t3, float* __restrict__ t4)
{
    const int lane = threadIdx.x & 31;
    const int half = lane >> 4;        // 0: K lo pair, 1: K hi pair
    const int l    = lane & 15;
    const int n0   = blockIdx.x * 16;  // 0..80
    const int m0   = blockIdx.y * 16;  // node tile

    v8f c1 = {}, c2 = {}, c3 = {}, c4 = {};
    const float* xrow = x + (size_t)(m0 + l) * D;

    for (int k0 = 0; k0 < D; k0 += 4) {
        const int ka = k0 + 2 * half;
        v2f a;  a.x = xrow[ka];  a.y = xrow[ka + 1];

        v2f f1, f2, f3, f4;
        f1.x = w1[ka * D + n0 + l]; f1.y = w1[(ka + 1) * D + n0 + l];
        f2.x = w2[ka * D + n0 + l]; f2.y = w2[(ka + 1) * D + n0 + l];
        f3.x = w3[ka * D + n0 + l]; f3.y = w3[(ka + 1) * D + n0 + l];
        f4.x = w4[ka * D + n0 + l]; f4.y = w4[(ka + 1) * D + n0 + l];

        c1 = __builtin_amdgcn_wmma_f32_16x16x4_f32(false, a, false, f1, (short)0, c1, false, false);
        c2 = __builtin_amdgcn_wmma_f32_16x16x4_f32(false, a, false, f2, (short)0, c2, false, false);
        c3 = __builtin_amdgcn_wmma_f32_16x16x4_f32(false, a, false, f3, (short)0, c3, false, false);
        c4 = __builtin_amdgcn_wmma_f32_16x16x4_f32(false, a, false, f4, (short)0, c4, false, false);
    }

    const int col = n0 + l;
    const float bb1 = b1[col], bb2 = b2[col], bb3 = b3[col], bb4 = b4[col];
#pragma unroll
    for (int r = 0; r < 8; ++r) {
        const size_t idx = (size_t)(m0 + r + 8 * half) * D + col;
        t1[idx] = c1[r] + bb1;
        t2[idx] = c2[r] + bb2;
        t3[idx] = c3[r] + bb3;
        t4[idx] = c4[r] + bb4;
    }
}

// -------- pass A: one wave per edge: score + segment max --------
__global__ void GNN_edge_score(const int* __restrict__ ei,
                               const float* __restrict__ eattr,
                               const float* __restrict__ t3,
                               const float* __restrict__ t4,
                               const float* __restrict__ w5row,
                               const float* __restrict__ b5,
                               float* __restrict__ score,
                               float* __restrict__ segmax)
{
    const int lane = threadIdx.x & 31;
    const long e = (long)blockIdx.x * (blockDim.x >> 5) + (threadIdx.x >> 5);
    if (e >= N_EDGES) return;
    const int src = ei[e];
    const int dst = ei[N_EDGES + e];
    const float ea = eattr[e];

    float partial = 0.0f;
#pragma unroll
    for (int r = 0; r < 3; ++r) {
        const int k = lane + 32 * r;
        const float bi = t4[(size_t)src * D + k] + ea * w5row[k] + b5[k];
        partial = fmaf(t3[(size_t)dst * D + k], bi, partial);
    }
#pragma unroll
    for (int m = 16; m >= 1; m >>= 1) partial += __shfl_xor(partial, m, 32);

    const float s = partial * 0.10206207261596575f;   // 1/sqrt(96)
    if (lane == 0) { score[e] = s; atomicMaxF(&segmax[dst], s); }
}

// -------- pass B: thread per edge: exp + segment sum --------
__global__ void GNN_edge_exp(const int* __restrict__ ei,
                             float* __restrict__ score,
                             const float* __restrict__ segmax,
                             float* __restrict__ denom)
{
    const long e = (long)blockIdx.x * blockDim.x + threadIdx.x;
    if (e >= N_EDGES) return;
    const int dst = ei[N_EDGES + e];
    const float ex = __expf(score[e] - segmax[dst]);
    score[e] = ex;
    atomicAdd(&denom[dst], ex);
}

// -------- pass C: one wave per edge: alpha * t2[src] scattered to aggr[dst] --------
__global__ void GNN_edge_scatter(const int* __restrict__ ei,
                                 const float* __restrict__ score,
                                 const float* __restrict__ denom,
                                 const float* __restrict__ t2,
                                 float* __restrict__ aggr)
{
    const int lane = threadIdx.x & 31;
    const long e = (long)blockIdx.x * (blockDim.x >> 5) + (threadIdx.x >> 5);
    if (e >= N_EDGES) return;
    const int src = ei[e];
    const int dst = ei[N_EDGES + e];
    const float alpha = score[e] / (denom[dst] + 1e-16f);
#pragma unroll
    for (int r = 0; r < 3; ++r) {
        const int k = lane + 32 * r;
        atomicAdd(&aggr[(size_t)dst * D + k], alpha * t2[(size_t)src * D + k]);
    }
}

// -------- pool: node_out = t1 + aggr; per-graph sums & counts --------
// 96 threads/block (one per feature), 64 nodes/block, register accumulation,
// flush on graph-id change (batch is sorted -> ~1 flush per block).
__global__ void GNN_pool(const float* __restrict__ t1,
                         const float* __restrict__ aggr,
                         const int* __restrict__ batch,
                         float* __restrict__ sums,
                         float* __restrict__ cnt)
{
    const int k  = threadIdx.x;           // 0..95
    const int i0 = blockIdx.x * 64;
    const int iend = min(i0 + 64, N_NODES);
    int cur = batch[i0];
    float acc = 0.0f, c = 0.0f;
    for (int i = i0; i < iend; ++i) {
        const int b = batch[i];
        if (b != cur) {
            atomicAdd(&sums[cur * D + k], acc);
            if (k == 0) atomicAdd(&cnt[cur], c);
            acc = 0.0f; c = 0.0f; cur = b;
        }
        acc += t1[(size_t)i * D + k] + aggr[(size_t)i * D + k];
        c += 1.0f;
    }
    atomicAdd(&sums[cur * D + k], acc);
    if (k == 0) atomicAdd(&cnt[cur], c);
}

// -------- finalize: mean pool --------
__global__ void GNN_finalize(const float* __restrict__ sums,
                             const float* __restrict__ cnt,
                             float* __restrict__ out)
{
    const int g = blockIdx.x, k = threadIdx.x;
    out[g * D + k] = sums[g * D + k] / fmaxf(cnt[g], 1.0f);
}

extern "C" void kernel_launch(void* const* d_in, const int* in_sizes, int n_in,
                              void* d_out, int out_size, void* d_ws, size_t ws_size,
                              hipStream_t stream) {
    (void)in_sizes; (void)n_in; (void)out_size; (void)ws_size;

    const float* x     = (const float*)d_in[0];
    const int*   ei    = (const int*)  d_in[1];
    const float* eattr = (const float*)d_in[2];
    const int*   batch = (const int*)  d_in[3];
    const float* w1w = (const float*)d_in[4],  *w1b = (const float*)d_in[5];
    const float* w2w = (const float*)d_in[6],  *w2b = (const float*)d_in[7];
    const float* w3w = (const float*)d_in[8],  *w3b = (const float*)d_in[9];
    const float* w4w = (const float*)d_in[10], *w4b = (const float*)d_in[11];
    const float* w5w = (const float*)d_in[12], *w5b = (const float*)d_in[13];
    float* out = (float*)d_out;

    // workspace layout (floats)
    float* ws = (float*)d_ws;
    float* t1     = ws;                         // 4,800,000
    float* t2     = t1 + (size_t)N_NODES * D;   // 4,800,000
    float* t3     = t2 + (size_t)N_NODES * D;
    float* t4     = t3 + (size_t)N_NODES * D;
    float* aggr   = t4 + (size_t)N_NODES * D;   // 4,800,000
    float* score  = aggr + (size_t)N_NODES * D; // 800,000
    float* segmax = score + N_EDGES;            // 50,000
    float* denom  = segmax + N_NODES;           // 50,000
    float* sums   = denom + N_NODES;            // 4,800
    float* cnt    = sums + N_GRAPHS * D;        // 50

    // 1) init
    {
        int threads = 256;
        int blocks = (N_NODES * D + threads - 1) / threads;
        GNN_init<<<blocks, threads, 0, stream>>>(segmax, denom, aggr, sums, cnt);
    }
    // 2) fused node GEMMs (WMMA f32)
    {
        dim3 grid(D / 16, N_NODES / 16);   // (6, 3125)
        GNN_gemm4_wmma<<<grid, 32, 0, stream>>>(x, w1w, w1b, w2w, w2b, w3w, w3b,
                                                w4w, w4b, t1, t2, t3, t4);
    }
    // 3) edge pass A: scores + segment max (wave per edge)
    {
        int threads = 256;                           // 8 waves/block
        int blocks = (N_EDGES + 7) / 8;
        GNN_edge_score<<<blocks, threads, 0, stream>>>(ei, eattr, t3, t4, w5w, w5b,
                                                       score, segmax);
    }
    // 4) edge pass B: exp + segment sum
    {
        int threads = 256;
        int blocks = (N_EDGES + threads - 1) / threads;
        GNN_edge_exp<<<blocks, threads, 0, stream>>>(ei, score, segmax, denom);
    }
    // 5) edge pass C: normalized message scatter
    {
        int threads = 256;
        int blocks = (N_EDGES + 7) / 8;
        GNN_edge_scatter<<<blocks, threads, 0, stream>>>(ei, score, denom, t2, aggr);
    }
    // 6) pool per graph
    {
        int blocks = (N_NODES + 63) / 64;
        GNN_pool<<<blocks, 96, 0, stream>>>(t1, aggr, batch, sums, cnt);
    }
    // 7) finalize mean
    GNN_finalize<<<N_GRAPHS, 96, 0, stream>>>(sums, cnt, out);
}